// Enhancement_11819749999257
// MI455X (gfx1250) — compile-verified
//
#include <hip/hip_runtime.h>

typedef float v2f __attribute__((ext_vector_type(2)));
typedef float v8f __attribute__((ext_vector_type(8)));

#define NJ_TOTAL 74          // 21 + 21 + 32
#define NPIX 49
#define CDIM 64

// ---------------------------------------------------------------------------
// Kernel 1: fused per-joint embedding GEMM
//   femb[col, :] = W2 @ relu( BN(W1 @ x_col) )  + b2     (BN folded into W1/b1)
// One block = 16 joint-columns, 128 threads = 4 waves.
// Wave w computes output rows [16w, 16w+16) via V_WMMA_F32_16X16X4_F32,
// chaining 16 k-steps (K=64).
// ---------------------------------------------------------------------------
__global__ __launch_bounds__(128) void embed_gemm_kernel(
    const float* __restrict__ feat,          // (B, 64, Jh)
    const float* __restrict__ w1, const float* __restrict__ b1,
    const float* __restrict__ bn_g, const float* __restrict__ bn_b,
    const float* __restrict__ bn_m, const float* __restrict__ bn_v,
    const float* __restrict__ w2, const float* __restrict__ b2,
    float* __restrict__ femb,                // (ncols, 64) for this head
    int Jh)
{
    __shared__ float s_w1[64][66];   // stride 66: even (8B-aligned b64 A-frag
    __shared__ float s_w2[64][66];   // loads) and bank-spread across lanes
    __shared__ float s_x [64][16];
    __shared__ float s_y1[64][16];
    __shared__ float s_b1p[64];
    __shared__ float s_b2[64];
    __shared__ float s_scale[64];

    const int t = threadIdx.x;

    // Fold BN into conv1 (eval mode, running stats)
    if (t < 64) {
        float s = bn_g[t] * rsqrtf(bn_v[t] + 1e-5f);
        s_scale[t] = s;
        s_b1p[t]   = (b1[t] - bn_m[t]) * s + bn_b[t];
        s_b2[t]    = b2[t];
    }
    __syncthreads();

    // Stage weights (W1 pre-scaled by BN) and the 64x16 X tile.
    for (int e = t; e < 64 * 64; e += 128) {
        int o = e >> 6, c = e & 63;
        s_w1[o][c] = w1[e] * s_scale[o];
        s_w2[o][c] = w2[e];
    }
    const int col0 = blockIdx.x * 16;
    for (int e = t; e < 64 * 16; e += 128) {
        int c = e >> 4, lc = e & 15;
        int col = col0 + lc;
        int b = col / Jh;
        int j = col - b * Jh;
        s_x[c][lc] = feat[((size_t)b * 64 + c) * Jh + j];
    }
    __syncthreads();

    const int lane = t & 31;
    const int wv   = t >> 5;
    const int mbase = wv * 16;
    const int hi = lane >> 4;       // lane half selects K+0/1 vs K+2/3 of A/B frag
    const int lm = lane & 15;       // A: row M  /  B: col N

    // ---- GEMM1: Y1 = relu(W1' @ X + b1') ----
    v8f acc = {};
    #pragma unroll
    for (int kb = 0; kb < 16; ++kb) {
        int k0 = kb * 4 + hi * 2;
        v2f a, bf;
        a[0]  = s_w1[mbase + lm][k0];
        a[1]  = s_w1[mbase + lm][k0 + 1];
        bf[0] = s_x[k0][lm];
        bf[1] = s_x[k0 + 1][lm];
        acc = __builtin_amdgcn_wmma_f32_16x16x4_f32(
            /*neg_a=*/false, a, /*neg_b=*/false, bf,
            /*c_mod=*/(short)0, acc, /*reuse_a=*/false, /*reuse_b=*/false);
    }
    // C/D layout: VGPR v holds M = rbase + v for this lane's half; N = lm.
    const int rbase = mbase + hi * 8;
    #pragma unroll
    for (int v = 0; v < 8; ++v) {
        float y = acc[v] + s_b1p[rbase + v];
        s_y1[rbase + v][lm] = fmaxf(y, 0.0f);
    }
    __syncthreads();

    // ---- GEMM2: Y2 = W2 @ Y1 + b2 ----
    v8f acc2 = {};
    #pragma unroll
    for (int kb = 0; kb < 16; ++kb) {
        int k0 = kb * 4 + hi * 2;
        v2f a, bf;
        a[0]  = s_w2[mbase + lm][k0];
        a[1]  = s_w2[mbase + lm][k0 + 1];
        bf[0] = s_y1[k0][lm];
        bf[1] = s_y1[k0 + 1][lm];
        acc2 = __builtin_amdgcn_wmma_f32_16x16x4_f32(
            false, a, false, bf, (short)0, acc2, false, false);
    }
    // Each lane owns 8 consecutive rows of one column -> 2 contiguous b128 stores.
    float* outp = femb + (size_t)(col0 + lm) * 64 + rbase;
    float4 o0, o1;
    o0.x = acc2[0] + s_b2[rbase + 0];
    o0.y = acc2[1] + s_b2[rbase + 1];
    o0.z = acc2[2] + s_b2[rbase + 2];
    o0.w = acc2[3] + s_b2[rbase + 3];
    o1.x = acc2[4] + s_b2[rbase + 4];
    o1.y = acc2[5] + s_b2[rbase + 5];
    o1.z = acc2[6] + s_b2[rbase + 6];
    o1.w = acc2[7] + s_b2[rbase + 7];
    *(float4*)(outp)     = o0;
    *(float4*)(outp + 4) = o1;
}

// ---------------------------------------------------------------------------
// Kernel 2: grid splat -> (B, 74*64, 7, 7). One block per (batch, joint),
// 256 threads = 8 waves; wave w owns channels [8w, 8w+8).
// Each lane holds its two grid weights (p=lane, p=lane+32) in registers;
// inner loop per channel row = 2 v_mul + 2 contiguous global_store_b32.
// 237 MB written exactly once, coalesced = HBM roofline of the problem.
// ---------------------------------------------------------------------------
__global__ __launch_bounds__(256) void proj_kernel(
    const float* __restrict__ j2d_r, const float* __restrict__ j2d_l,
    const float* __restrict__ kp2d_o,
    const float* __restrict__ femb,          // (18944, 64): r | l | o blocks
    float* __restrict__ out)
{
    __shared__ float s_f[64];

    const int bid = blockIdx.x;
    const int b  = bid / NJ_TOTAL;
    const int jj = bid - b * NJ_TOTAL;

    const float* uvsrc;
    int Jh, jh, headoff;
    if (jj < 21)      { uvsrc = j2d_r;  Jh = 21; jh = jj;      headoff = 0;        }
    else if (jj < 42) { uvsrc = j2d_l;  Jh = 21; jh = jj - 21; headoff = 256 * 21; }
    else              { uvsrc = kp2d_o; Jh = 32; jh = jj - 42; headoff = 256 * 42; }

    const int t    = threadIdx.x;
    const int lane = t & 31;
    const int wv   = t >> 5;

    const float* frow = femb + (size_t)(headoff + b * Jh + jh) * 64;
    if (t < 64) s_f[t] = frow[t];

    // uv = (j2d+1)/2 * 7 ; grid[p] = ((p%7)+0.5, (p/7)+0.5)
    const float u0 = (uvsrc[(b * Jh + jh) * 2 + 0] + 1.0f) * 3.5f;
    const float u1 = (uvsrc[(b * Jh + jh) * 2 + 1] + 1.0f) * 3.5f;

    // Per-lane register weights for p = lane and p = lane + 32.
    int p0 = lane, p1 = lane + 32;
    float dx0 = (float)(p0 % 7) + 0.5f - u0 + 1e-6f;
    float dy0 = (float)(p0 / 7) + 0.5f - u1 + 1e-6f;
    float w0  = 1.0f - fminf(sqrtf(dx0 * dx0 + dy0 * dy0), 1.0f);
    float dx1 = (float)(p1 % 7) + 0.5f - u0 + 1e-6f;
    float dy1 = (float)(p1 / 7) + 0.5f - u1 + 1e-6f;
    float w1  = 1.0f - fminf(sqrtf(dx1 * dx1 + dy1 * dy1), 1.0f);

    __syncthreads();

    // Two uniform-address b128 LDS loads give this wave's 8 channel values.
    const float4 fa = *(const float4*)&s_f[wv * 8];
    const float4 fb = *(const float4*)&s_f[wv * 8 + 4];
    const float fv[8] = { fa.x, fa.y, fa.z, fa.w, fb.x, fb.y, fb.z, fb.w };

    float* ob = out + (size_t)b * (NJ_TOTAL * CDIM * NPIX)
                    + (size_t)jj * (CDIM * NPIX)
                    + (size_t)(wv * 8) * NPIX;
    #pragma unroll
    for (int i = 0; i < 8; ++i) {
        float f = fv[i];
        float* row = ob + i * NPIX;
        row[lane] = f * w0;                       // p = 0..31
        if (lane < NPIX - 32) row[p1] = f * w1;   // p = 32..48 (EXEC-masked)
    }
}

// ---------------------------------------------------------------------------
extern "C" void kernel_launch(void* const* d_in, const int* in_sizes, int n_in,
                              void* d_out, int out_size, void* d_ws, size_t ws_size,
                              hipStream_t stream) {
    const float* j2d_r  = (const float*)d_in[0];
    const float* j2d_l  = (const float*)d_in[1];
    const float* kp2d_o = (const float*)d_in[2];
    const float* feat_r = (const float*)d_in[3];
    const float* feat_l = (const float*)d_in[4];
    const float* feat_o = (const float*)d_in[5];
    const float* w1     = (const float*)d_in[6];
    const float* b1     = (const float*)d_in[7];
    const float* bn_g   = (const float*)d_in[8];
    const float* bn_b   = (const float*)d_in[9];
    const float* bn_m   = (const float*)d_in[10];
    const float* bn_v   = (const float*)d_in[11];
    const float* w2     = (const float*)d_in[12];
    const float* b2     = (const float*)d_in[13];
    float* out  = (float*)d_out;
    float* femb = (float*)d_ws;              // needs 18944*64*4 = ~4.85 MB

    // Columns per head: 256*21=5376, 5376, 256*32=8192 (all /16 exact).
    embed_gemm_kernel<<<5376 / 16, 128, 0, stream>>>(
        feat_r, w1, b1, bn_g, bn_b, bn_m, bn_v, w2, b2, femb, 21);
    embed_gemm_kernel<<<5376 / 16, 128, 0, stream>>>(
        feat_l, w1, b1, bn_g, bn_b, bn_m, bn_v, w2, b2, femb + (size_t)5376 * 64, 21);
    embed_gemm_kernel<<<8192 / 16, 128, 0, stream>>>(
        feat_o, w1, b1, bn_g, bn_b, bn_m, bn_v, w2, b2, femb + (size_t)10752 * 64, 32);

    proj_kernel<<<256 * NJ_TOTAL, 256, 0, stream>>>(j2d_r, j2d_l, kp2d_o, femb, out);
}